// SpatialAttentionGCN_85487029060117
// MI455X (gfx1250) — compile-verified
//
#include <hip/hip_runtime.h>
#include <hip/hip_bf16.h>
#include <math.h>

// ---------------------------------------------------------------------------
// SpatialAttentionGCN on MI455X (gfx1250), fp32 end-to-end via
// v_wmma_f32_16x16x4_f32. Problem is L2-resident (~25MB << 192MB L2) so we
// keep full fp32 precision, lean on the CDNA5 matrix pipe, and use the
// Tensor Data Mover for double-buffered global->LDS staging in the graph
// matmuls (the kernel that runs 4x).
// ---------------------------------------------------------------------------

typedef float v2f __attribute__((ext_vector_type(2)));
typedef float v8f __attribute__((ext_vector_type(8)));
typedef unsigned int u32x4 __attribute__((ext_vector_type(4)));
typedef int i32x4 __attribute__((ext_vector_type(4)));
typedef int i32x8 __attribute__((ext_vector_type(8)));

#define WMMA4(a, b, c) \
    __builtin_amdgcn_wmma_f32_16x16x4_f32(false, (a), false, (b), (short)0, (c), false, false)

#define NVERT 512
#define NDIM 64
#define NBL 128          // B*L
#define NFEAT 32768      // NVERT*NDIM
#define NDM 128
#define NEG_SLOPE 0.01f
#define NSPLITK 32
#define TPLD 72          // padded LDS row stride for the 512x64 tanh stripe

// workspace layout (float offsets)
#define OFF_COE 0ull
#define OFF_H   16ull
#define OFF_TA  (OFF_H  + 4194304ull)   // Tx1 / Tx3 / later G
#define OFF_TB  (OFF_TA + 4194304ull)   // Tx2 / later S_x
#define OFF_AX  (OFF_TB + 4194304ull)
#define OFF_KX  (OFF_AX + 4194304ull)
#define OFF_PART (OFF_KX + 4194304ull)  // NSPLITK*128*128 split-K partials

__device__ __forceinline__ float leaky(float v) {
    return v >= 0.f ? v : NEG_SLOPE * v;
}

// --- TDM: DMA one contiguous 4096-float (16KB) chunk global -> LDS ---------
// This toolchain exposes the 6-arg builtin (g0, g1, g2, g3, g4, cpol).
__device__ __forceinline__ void tdm_load_chunk(const float* gptr, unsigned lds_off) {
    unsigned long long ga = (unsigned long long)(uintptr_t)gptr;
    u32x4 g0 = {};
    g0[0] = 1u;                                      // count=1 valid user D#
    g0[1] = lds_off;                                 // lds_addr (bytes)
    g0[2] = (unsigned)ga;                            // global_addr[31:0]
    g0[3] = ((unsigned)(ga >> 32) & 0x01FFFFFFu)     // global_addr[56:32]
            | (2u << 30);                            // type = 2 ("image")
    i32x8 g1 = {};
    g1[0] = 2 << 16;                                 // data_size = 4 bytes
    g1[1] = (int)(4096u << 16);                      // tensor_dim0 = 4096
    g1[2] = (int)(1u << 16);                         // tensor_dim1 = 1
    g1[3] = (int)(4096u << 16);                      // tile_dim0 = 4096
    g1[4] = 0;                                       // tile_dim1/2 unused
    g1[5] = 4096;                                    // tensor_dim0_stride
    g1[6] = 0;
    g1[7] = 0;
    i32x4 gz = {};
    i32x8 gz8 = {};
    __builtin_amdgcn_tensor_load_to_lds(g0, g1, gz, gz, gz8, 0);
}

// --- Chebyshev coefficients from relu(temp), K=3 ----------------------------
__global__ void k_coe(const float* __restrict__ temp, float* __restrict__ coe) {
    if (threadIdx.x == 0 && blockIdx.x == 0) {
        const float xs[4] = {-0.9238795325f, -0.3826834324f,
                              0.3826834324f,  0.9238795325f};
        float c[4] = {0.f, 0.f, 0.f, 0.f};
        for (int j = 0; j < 4; ++j) {
            float x = xs[j];
            float t = fmaxf(temp[j], 0.f);
            float Tv[4];
            Tv[0] = 1.f;
            Tv[1] = x;
            Tv[2] = 2.f * x * x - 1.f;
            Tv[3] = (4.f * x * x - 3.f) * x;
            for (int i = 0; i < 4; ++i) c[i] += Tv[i] * t;
        }
        for (int i = 0; i < 4; ++i) coe[i] = 0.5f * c[i];  // 2/(K+1) = 0.5
    }
}

// --- h = x ⊗ Wp + bp ; h[bl,n,d] ------------------------------------------
__global__ void k_h(const float* __restrict__ x, const float* __restrict__ Wp,
                    const float* __restrict__ bp, float* __restrict__ H) {
    unsigned idx = blockIdx.x * 256u + threadIdx.x;   // 128*512*64 total
    unsigned d = idx & 63u;
    H[idx] = x[idx >> 6] * Wp[d] + bp[d];
}

// --- Y[bl] = A(512x512) @ X[bl](512x64), fused Cheb epilogues --------------
// X is staged into LDS in 64-row chunks by the Tensor Data Mover (wave 0),
// double buffered so the DMA overlaps the WMMA stream.
// mode 0: Y = acc                          (Kx = Ksa @ h)
// mode 1: Y = acc; Ax = c0/2*H + c1*Y      (Tx1 init)
// mode 2: Y = 2*acc - Z; Ax += coe[cidx]*Y (Cheb step)
__global__ __launch_bounds__(128) void
k_graphmm(const float* __restrict__ A, const float* __restrict__ X,
          const float* __restrict__ Z, float* __restrict__ Y,
          float* __restrict__ Ax, const float* __restrict__ H,
          const float* __restrict__ coe, int cidx, int mode) {
    __shared__ __align__(16) float Xs[2][4096];      // 2 x 16KB chunks
    const int wave = threadIdx.x >> 5, lane = threadIdx.x & 31;
    const int lo = lane & 15, hi = lane >> 4;
    const int bl = blockIdx.x >> 3;
    const int mt = (blockIdx.x & 7) * 4 + wave;      // 32 row-tiles of 16
    const int row0 = mt * 16;
    const float* Xb = X + (size_t)bl * NFEAT;
    const unsigned lds0 = (unsigned)(uintptr_t)&Xs[0][0];
    const unsigned lds1 = (unsigned)(uintptr_t)&Xs[1][0];

    v8f acc[4] = {};
    if (wave == 0) tdm_load_chunk(Xb, lds0);
    for (int kc = 0; kc < 8; ++kc) {
        if (wave == 0) {
            if (kc < 7) {
                tdm_load_chunk(Xb + (size_t)(kc + 1) * 4096,
                               ((kc + 1) & 1) ? lds1 : lds0);
                __builtin_amdgcn_s_wait_tensorcnt(1);   // chunk kc landed
            } else {
                __builtin_amdgcn_s_wait_tensorcnt(0);
            }
        }
        __syncthreads();
        const float* Xc = (kc & 1) ? &Xs[1][0] : &Xs[0][0];
        for (int k4 = 0; k4 < 64; k4 += 4) {
            const int ksl = k4 + 2 * hi;             // k within chunk
            const int kg = kc * 64 + ksl;            // global k (for A)
            v2f a = *(const v2f*)(A + (size_t)(row0 + lo) * NVERT + kg);
#pragma unroll
            for (int dt = 0; dt < 4; ++dt) {
                v2f b;
                b.x = Xc[ksl * NDIM + dt * 16 + lo];
                b.y = Xc[(ksl + 1) * NDIM + dt * 16 + lo];
                acc[dt] = WMMA4(a, b, acc[dt]);
            }
        }
        __syncthreads();                             // protect next overwrite
    }

    float cv = 0.f, c0h = 0.f;
    if (mode == 1) { c0h = coe[0] * 0.5f; cv = coe[1]; }
    else if (mode == 2) { cv = coe[cidx]; }

#pragma unroll
    for (int dt = 0; dt < 4; ++dt)
#pragma unroll
        for (int r = 0; r < 8; ++r) {
            int row = row0 + r + 8 * hi;
            int col = dt * 16 + lo;
            size_t idx = (size_t)bl * NFEAT + (size_t)row * NDIM + col;
            float av = acc[dt][r];
            if (mode == 0) {
                Y[idx] = av;
            } else if (mode == 1) {
                Y[idx] = av;
                Ax[idx] = c0h * H[idx] + cv * av;
            } else {
                float y = 2.f * av - Z[idx];
                Y[idx] = y;
                Ax[idx] += cv * y;
            }
        }
}

// --- flash-style spatial attention: S_x = softmax(Vsa @ tanh(h Kx^T + b)) @ h
// One workgroup (32 waves) per bl slice; streams 64-column stripes of the
// never-materialized 512x512 score matrix. The tanh stripe (512x64, padded
// to stride 72 so the two 16-lane halves hit disjoint LDS banks) lives in
// dynamic LDS; one Vsa A-fragment feeds 4 score accumulators.
__global__ __launch_bounds__(1024) void
k_attn(const float* __restrict__ H, const float* __restrict__ Kx,
       const float* __restrict__ Vsa, const float* __restrict__ bsa,
       float* __restrict__ Sx) {
    extern __shared__ float tP[];                    // 512*TPLD floats (144KB)
    const int bl = blockIdx.x;
    const int wave = threadIdx.x >> 5, lane = threadIdx.x & 31;
    const int lo = lane & 15, hi = lane >> 4;
    const int n0 = wave * 16;                        // this wave's output rows
    const float* Hb  = H  + (size_t)bl * NFEAT;
    const float* Kxb = Kx + (size_t)bl * NFEAT;

    float Mrow[8], Lrow[8];
    v8f O[4] = {};
#pragma unroll
    for (int r = 0; r < 8; ++r) { Mrow[r] = -1e30f; Lrow[r] = 0.f; }

    for (int ct = 0; ct < 8; ++ct) {
        const int cg = ct * 64;
        // Phase A: tP[m, 0..63] = tanh(h[m,:] . Kx[cg+c,:] + bsa[m,cg+c])
#pragma unroll
        for (int ci = 0; ci < 4; ++ci) {
            v8f t = {};
            for (int k0 = 0; k0 < NDIM; k0 += 4) {
                const int ks = k0 + 2 * hi;
                v2f a = *(const v2f*)(Hb + (size_t)(n0 + lo) * NDIM + ks);
                v2f b = *(const v2f*)(Kxb + (size_t)(cg + ci * 16 + lo) * NDIM + ks);
                t = WMMA4(a, b, t);
            }
#pragma unroll
            for (int r = 0; r < 8; ++r) {
                int m = n0 + r + 8 * hi;
                tP[m * TPLD + ci * 16 + lo] =
                    tanhf(t[r] + bsa[(size_t)m * NVERT + cg + ci * 16 + lo]);
            }
        }
        __syncthreads();
        // Phase B1: 4 score tiles; one Vsa fragment feeds all 4
        v8f s[4] = {};
        for (int k0 = 0; k0 < NVERT; k0 += 4) {
            const int ks = k0 + 2 * hi;
            v2f a = *(const v2f*)(Vsa + (size_t)(n0 + lo) * NVERT + ks);
#pragma unroll
            for (int ci = 0; ci < 4; ++ci) {
                v2f b;
                b.x = tP[ks * TPLD + ci * 16 + lo];
                b.y = tP[(ks + 1) * TPLD + ci * 16 + lo];
                s[ci] = WMMA4(a, b, s[ci]);
            }
        }
        __syncthreads();                 // all waves done reading tanh stripe
        // Per 16-col subtile: online softmax + O += P @ h (P relayout through
        // this wave's private tP slice; intra-wave DS ordering via fence).
#pragma unroll
        for (int ci = 0; ci < 4; ++ci) {
            float pv[8];
#pragma unroll
            for (int r = 0; r < 8; ++r) {
                float v = s[ci][r];
                float rm = v;                                // row max
                rm = fmaxf(rm, __shfl_xor(rm, 1, 32));
                rm = fmaxf(rm, __shfl_xor(rm, 2, 32));
                rm = fmaxf(rm, __shfl_xor(rm, 4, 32));
                rm = fmaxf(rm, __shfl_xor(rm, 8, 32));
                float mnew = fmaxf(Mrow[r], rm);
                float p = __expf(v - mnew);
                float rs = p;                                // row sum
                rs += __shfl_xor(rs, 1, 32);
                rs += __shfl_xor(rs, 2, 32);
                rs += __shfl_xor(rs, 4, 32);
                rs += __shfl_xor(rs, 8, 32);
                float corr = __expf(Mrow[r] - mnew);
                Lrow[r] = Lrow[r] * corr + rs;
                Mrow[r] = mnew;
                pv[r] = p;
#pragma unroll
                for (int dt = 0; dt < 4; ++dt) O[dt][r] *= corr;
            }
#pragma unroll
            for (int r = 0; r < 8; ++r)
                tP[(n0 + r + 8 * hi) * TPLD + ci * 16 + lo] = pv[r];
            __threadfence_block();
#pragma unroll
            for (int kc = 0; kc < 4; ++kc) {
                const int kk = kc * 4 + 2 * hi;
                v2f a = *(const v2f*)(&tP[(n0 + lo) * TPLD + ci * 16 + kk]);
#pragma unroll
                for (int dt = 0; dt < 4; ++dt) {
                    v2f b;
                    b.x = Hb[(size_t)(cg + ci * 16 + kk) * NDIM + dt * 16 + lo];
                    b.y = Hb[(size_t)(cg + ci * 16 + kk + 1) * NDIM + dt * 16 + lo];
                    O[dt] = WMMA4(a, b, O[dt]);
                }
            }
        }
        __syncthreads();                 // before next stripe overwrites tP
    }
    // normalize and write S_x
#pragma unroll
    for (int dt = 0; dt < 4; ++dt)
#pragma unroll
        for (int r = 0; r < 8; ++r) {
            int row = n0 + r + 8 * hi;
            int col = dt * 16 + lo;
            Sx[(size_t)bl * NFEAT + (size_t)row * NDIM + col] = O[dt][r] / Lrow[r];
        }
}

// --- G = leaky((Ax + Sx) @ W1 + b1) ----------------------------------------
__global__ __launch_bounds__(128) void
k_theta(const float* __restrict__ Ax, const float* __restrict__ Sx,
        const float* __restrict__ W1, const float* __restrict__ b1,
        float* __restrict__ G) {
    const int wave = threadIdx.x >> 5, lane = threadIdx.x & 31;
    const int lo = lane & 15, hi = lane >> 4;
    const int bl = blockIdx.x >> 3;
    const int row0 = ((blockIdx.x & 7) * 4 + wave) * 16;

    v8f acc[4] = {};
    for (int k0 = 0; k0 < NDIM; k0 += 4) {
        const int ks = k0 + 2 * hi;
        size_t ai = (size_t)bl * NFEAT + (size_t)(row0 + lo) * NDIM + ks;
        v2f a1 = *(const v2f*)(Ax + ai);
        v2f a2 = *(const v2f*)(Sx + ai);
        v2f a = a1 + a2;
#pragma unroll
        for (int dt = 0; dt < 4; ++dt) {
            v2f b;
            b.x = W1[(size_t)ks * NDIM + dt * 16 + lo];
            b.y = W1[(size_t)(ks + 1) * NDIM + dt * 16 + lo];
            acc[dt] = WMMA4(a, b, acc[dt]);
        }
    }
#pragma unroll
    for (int dt = 0; dt < 4; ++dt)
#pragma unroll
        for (int r = 0; r < 8; ++r) {
            int row = row0 + r + 8 * hi;
            int col = dt * 16 + lo;
            G[(size_t)bl * NFEAT + (size_t)row * NDIM + col] =
                leaky(acc[dt][r] + b1[col]);
        }
}

// --- final GEMM (128 x 32768) @ (32768 x 128), split-K=32, deterministic ---
__global__ __launch_bounds__(128) void
k_lin_part(const float* __restrict__ G, const float* __restrict__ Wl,
           float* __restrict__ part) {
    const int wave = threadIdx.x >> 5, lane = threadIdx.x & 31;
    const int lo = lane & 15, hi = lane >> 4;
    const int gw = blockIdx.x * 4 + wave;   // 2048 waves
    const int ks = gw >> 6;
    const int mt = (gw & 63) >> 3, nt = gw & 7;
    const int row0 = mt * 16, col0 = nt * 16;
    const int kend = (ks + 1) * 1024;

    v8f acc = {};
    for (int k0 = ks * 1024; k0 < kend; k0 += 4) {
        const int kk = k0 + 2 * hi;
        v2f a = *(const v2f*)(G + (size_t)(row0 + lo) * NFEAT + kk);
        v2f b;
        b.x = Wl[(size_t)kk * NDM + col0 + lo];
        b.y = Wl[(size_t)(kk + 1) * NDM + col0 + lo];
        acc = WMMA4(a, b, acc);
    }
#pragma unroll
    for (int r = 0; r < 8; ++r) {
        int row = row0 + r + 8 * hi;
        part[((size_t)ks * NBL + row) * NDM + col0 + lo] = acc[r];
    }
}

__global__ void k_lin_red(const float* __restrict__ part,
                          const float* __restrict__ blb,
                          float* __restrict__ out) {
    unsigned i = blockIdx.x * 256u + threadIdx.x;   // 128*128
    unsigned col = i & 127u;
    float s = 0.f;
#pragma unroll
    for (int ks = 0; ks < NSPLITK; ++ks) s += part[(size_t)ks * NBL * NDM + i];
    out[i] = leaky(s + blb[col]);
}

// ---------------------------------------------------------------------------
extern "C" void kernel_launch(void* const* d_in, const int* in_sizes, int n_in,
                              void* d_out, int out_size, void* d_ws, size_t ws_size,
                              hipStream_t stream) {
    (void)in_sizes; (void)n_in; (void)out_size; (void)ws_size;
    const float* x    = (const float*)d_in[0];
    const float* adj  = (const float*)d_in[1];
    const float* Wp   = (const float*)d_in[2];
    const float* bp   = (const float*)d_in[3];
    const float* Ksa  = (const float*)d_in[4];
    const float* Vsa  = (const float*)d_in[5];
    const float* bsa  = (const float*)d_in[6];
    const float* temp = (const float*)d_in[7];
    const float* W1   = (const float*)d_in[8];
    const float* b1   = (const float*)d_in[9];
    const float* Wl   = (const float*)d_in[10];
    const float* blb  = (const float*)d_in[11];
    float* out = (float*)d_out;

    float* ws   = (float*)d_ws;         // needs ~86 MB
    float* coe  = ws + OFF_COE;
    float* H    = ws + OFF_H;
    float* TA   = ws + OFF_TA;
    float* TB   = ws + OFF_TB;
    float* AX   = ws + OFF_AX;
    float* KX   = ws + OFF_KX;
    float* PART = ws + OFF_PART;

    k_coe<<<1, 32, 0, stream>>>(temp, coe);
    k_h<<<16384, 256, 0, stream>>>(x, Wp, bp, H);
    // Kx = Ksa @ h
    k_graphmm<<<1024, 128, 0, stream>>>(Ksa, H, nullptr, KX, nullptr, nullptr, coe, 0, 0);
    // Tx1 = adj @ h ; Ax = c0/2*h + c1*Tx1
    k_graphmm<<<1024, 128, 0, stream>>>(adj, H, nullptr, TA, AX, H, coe, 1, 1);
    // Tx2 = 2*adj@Tx1 - h ; Ax += c2*Tx2
    k_graphmm<<<1024, 128, 0, stream>>>(adj, TA, H, TB, AX, nullptr, coe, 2, 2);
    // Tx3 = 2*adj@Tx2 - Tx1 ; Ax += c3*Tx3   (in-place element overwrite is safe)
    k_graphmm<<<1024, 128, 0, stream>>>(adj, TB, TA, TA, AX, nullptr, coe, 3, 2);
    // S_x (flash attention) -> TB ; 512*TPLD floats of dynamic LDS
    k_attn<<<128, 1024, NVERT * TPLD * sizeof(float), stream>>>(H, KX, Vsa, bsa, TB);
    // G = leaky((Ax + Sx) @ W1 + b1) -> TA
    k_theta<<<1024, 128, 0, stream>>>(AX, TB, W1, b1, TA);
    // out = leaky(G @ Wl + bl)
    k_lin_part<<<512, 128, 0, stream>>>(TA, Wl, PART);
    k_lin_red<<<64, 256, 0, stream>>>(PART, blb, out);
}